// SSDLoss_40690520162376
// MI455X (gfx1250) — compile-verified
//
#include <hip/hip_runtime.h>
#include <stdint.h>

#define BB 64
#define DD 8732
#define CC 81
#define AA 24
#define MATCH_THRESH 0.5f
#define NEG_RATIO 3

#define ROWS_PER_BLOCK 128
#define TILE_DWORDS (ROWS_PER_BLOCK * CC)               // 10368
#define B128_ITERS  (TILE_DWORDS / (ROWS_PER_BLOCK*4))  // 20 full 16B/lane steps
#define TAIL_BASE   (B128_ITERS * ROWS_PER_BLOCK * 4)   // 10240 (tail = 128 dwords)
#define USE_ASYNC_LDS 1

// ---------------------------------------------------------------------------
// Kernel 1: matching (single workgroup). Computes, for the shared target set:
//   matched[d], bwa[d] (=box_with_annotation), labels[d], trueoff[d][4], N, Nf
// ---------------------------------------------------------------------------
__global__ __launch_bounds__(256) void match_kernel(
    const float* __restrict__ def_boxes,   // [D][4] cxcywh
    const float* __restrict__ targets,     // [B*A][6]
    int*   __restrict__ labels,            // [D]
    int*   __restrict__ matched,           // [D]
    int*   __restrict__ bwa,               // [D]
    float* __restrict__ trueoff,           // [D][4]
    float* __restrict__ meta)              // [0]=Nf, int slot [1]=N
{
    __shared__ float s_ann_pt[AA][4];      // lx,ly,rx,ry
    __shared__ float s_ann_area[AA];
    __shared__ float s_ann_c[AA][4];       // cx,cy,w,h
    __shared__ int   s_ann_cls[AA];
    __shared__ float s_biou[AA][256];      // per-annotation best iou per thread
    __shared__ int   s_bd[AA][256];        // per-annotation best d per thread
    __shared__ int   s_awb[AA];            // annotation_with_box
    __shared__ int   s_red[256];

    const int t = threadIdx.x;

    if (t < AA) {
        s_ann_cls[t] = (int)targets[t * 6 + 1];
        float cx = targets[t * 6 + 2], cy = targets[t * 6 + 3];
        float w  = targets[t * 6 + 4], h  = targets[t * 6 + 5];
        s_ann_c[t][0] = cx; s_ann_c[t][1] = cy; s_ann_c[t][2] = w; s_ann_c[t][3] = h;
        float lx = fmaxf(cx - 0.5f * w, 0.0f);
        float ly = fmaxf(cy - 0.5f * h, 0.0f);
        float rx = fminf(cx + 0.5f * w, 1.0f);
        float ry = fminf(cy + 0.5f * h, 1.0f);
        s_ann_pt[t][0] = lx; s_ann_pt[t][1] = ly; s_ann_pt[t][2] = rx; s_ann_pt[t][3] = ry;
        s_ann_area[t] = fmaxf(rx - lx, 0.0f) * fmaxf(ry - ly, 0.0f);
    }
    for (int a = 0; a < AA; ++a) { s_biou[a][t] = -2.0f; s_bd[a][t] = 0x7fffffff; }
    __syncthreads();

    for (int d = t; d < DD; d += 256) {
        float cx = def_boxes[d * 4 + 0], cy = def_boxes[d * 4 + 1];
        float w  = def_boxes[d * 4 + 2], h  = def_boxes[d * 4 + 3];
        float dlx = fmaxf(cx - 0.5f * w, 0.0f);
        float dly = fmaxf(cy - 0.5f * h, 0.0f);
        float drx = fminf(cx + 0.5f * w, 1.0f);
        float dry = fminf(cy + 0.5f * h, 1.0f);
        float area_d = fmaxf(drx - dlx, 0.0f) * fmaxf(dry - dly, 0.0f);

        float bmax = -1e30f; int barg = 0;
#pragma unroll
        for (int a = 0; a < AA; ++a) {
            float ltx = fmaxf(s_ann_pt[a][0], dlx);
            float lty = fmaxf(s_ann_pt[a][1], dly);
            float rbx = fminf(s_ann_pt[a][2], drx);
            float rby = fminf(s_ann_pt[a][3], dry);
            float inter = fmaxf(rbx - ltx, 0.0f) * fmaxf(rby - lty, 0.0f);
            float uni   = s_ann_area[a] + area_d - inter;
            float iou   = inter / uni;
            if (iou > bmax) { bmax = iou; barg = a; }          // first max (smallest a)
            if (iou > s_biou[a][t]) { s_biou[a][t] = iou; s_bd[a][t] = d; } // smallest d on tie
        }
        matched[d] = (bmax >= MATCH_THRESH) ? 1 : 0;
        bwa[d] = barg;
    }
    __syncthreads();

    // per-annotation reduce across the 256 thread-local bests (tie -> smaller d)
    if (t < AA) {
        float best = -2.0f; int bd = 0x7fffffff;
        for (int j = 0; j < 256; ++j) {
            float v = s_biou[t][j]; int dd = s_bd[t][j];
            if (v > best || (v == best && dd < bd)) { best = v; bd = dd; }
        }
        s_awb[t] = bd;
    }
    __syncthreads();

    // forced matches (sequential a-order; last annotation wins on conflicts)
    if (t == 0) {
        for (int a = 0; a < AA; ++a) {
            int dd = s_awb[a];
            matched[dd] = 1;
            bwa[dd] = a;
        }
    }
    __syncthreads();

    // N = matched.sum()
    int cnt = 0;
    for (int d = t; d < DD; d += 256) cnt += matched[d];
    s_red[t] = cnt; __syncthreads();
    for (int s = 128; s > 0; s >>= 1) { if (t < s) s_red[t] += s_red[t + s]; __syncthreads(); }
    if (t == 0) { meta[0] = (float)s_red[0]; ((int*)meta)[1] = s_red[0]; }

    // labels + true offsets
    for (int d = t; d < DD; d += 256) {
        if (matched[d]) {
            int a = bwa[d];
            labels[d] = s_ann_cls[a];
            float dcx = def_boxes[d * 4 + 0], dcy = def_boxes[d * 4 + 1];
            float dw  = def_boxes[d * 4 + 2], dh  = def_boxes[d * 4 + 3];
            float mcx = s_ann_c[a][0], mcy = s_ann_c[a][1];
            float mw  = s_ann_c[a][2], mh  = s_ann_c[a][3];
            trueoff[d * 4 + 0] = (mcx - dcx) / (dw * 0.1f);
            trueoff[d * 4 + 1] = (mcy - dcy) / (dh * 0.1f);
            trueoff[d * 4 + 2] = logf(mw / dw) / 0.2f;
            trueoff[d * 4 + 3] = logf(mh / dh) / 0.2f;
        } else {
            labels[d] = 0;
            trueoff[d * 4 + 0] = 0.f; trueoff[d * 4 + 1] = 0.f;
            trueoff[d * 4 + 2] = 0.f; trueoff[d * 4 + 3] = 0.f;
        }
    }
}

// ---------------------------------------------------------------------------
// Kernel 2: per-row cross-entropy + smooth-L1. One thread per (b,d) row,
// 128 rows staged per block into LDS via gfx1250 async global->LDS DMA
// (b128 bulk + b32 tail), non-temporal: the 181MB logit stream is read once
// and must not evict the 192MB L2.
// ---------------------------------------------------------------------------
__global__ __launch_bounds__(ROWS_PER_BLOCK) void ce_kernel(
    const float* __restrict__ pc,        // [B][D][C]
    const float* __restrict__ po,        // [B][D][4]
    const int*   __restrict__ labels,
    const int*   __restrict__ matched,
    const float* __restrict__ trueoff,
    float* __restrict__ negbuf,          // [B*D] ce per row
    float* __restrict__ regbuf)          // [B*D] reg per row (0 if unmatched)
{
    __shared__ float sh[TILE_DWORDS];
    const size_t base = (size_t)blockIdx.x * TILE_DWORDS;
    const float* __restrict__ src = pc + base;

#if USE_ASYNC_LDS
    // gfx1250 async global->LDS path, tracked by ASYNCcnt. 16B per lane per op.
#pragma unroll
    for (int it = 0; it < B128_ITERS; ++it) {
        const int i = it * (ROWS_PER_BLOCK * 4) + threadIdx.x * 4;   // 16B aligned
        unsigned lds_off = (unsigned)(uintptr_t)(&sh[i]);
        const float* g = src + i;
        asm volatile("global_load_async_to_lds_b128 %0, %1, off th:TH_LOAD_NT"
                     :: "v"(lds_off), "v"(g) : "memory");
    }
    {   // tail: TILE_DWORDS - TAIL_BASE == ROWS_PER_BLOCK dwords
        const int i = TAIL_BASE + threadIdx.x;
        unsigned lds_off = (unsigned)(uintptr_t)(&sh[i]);
        const float* g = src + i;
        asm volatile("global_load_async_to_lds_b32 %0, %1, off th:TH_LOAD_NT"
                     :: "v"(lds_off), "v"(g) : "memory");
    }
    asm volatile("s_wait_asynccnt 0x0" ::: "memory");
#else
    for (int i = threadIdx.x; i < TILE_DWORDS; i += ROWS_PER_BLOCK) sh[i] = src[i];
#endif
    __syncthreads();

    const int row = blockIdx.x * ROWS_PER_BLOCK + threadIdx.x;
    const int b = row / DD;
    const int d = row - b * DD;
    const float* __restrict__ x = &sh[threadIdx.x * CC];

    float m = x[0];
#pragma unroll 8
    for (int j = 1; j < CC; ++j) m = fmaxf(m, x[j]);
    float s = 0.0f;
#pragma unroll 8
    for (int j = 0; j < CC; ++j) s += __expf(x[j] - m);
    const int lab = labels[d];
    const float ce = m + __logf(s) - x[lab];      // lse - x[label]  (>= 0)
    __builtin_nontemporal_store(ce, &negbuf[row]);

    float r = 0.0f;
    if (matched[d]) {
        const float* o = po + (size_t)row * 4;
#pragma unroll
        for (int k = 0; k < 4; ++k) {
            float tdiff = __builtin_nontemporal_load(&o[k]) - trueoff[d * 4 + k];
            float a = fabsf(tdiff);
            r += (a < 1.0f) ? 0.5f * tdiff * tdiff : a - 0.5f;
        }
    }
    __builtin_nontemporal_store(r, &regbuf[row]);
}

// ---------------------------------------------------------------------------
// Kernel 3: per-image reductions + exact top-K hard-negative mining.
// One block per image. Keys live in LDS; 32-round bitwise kth-largest search.
// Deterministic: integer LDS atomics + float tree reductions only.
// ---------------------------------------------------------------------------
__device__ __forceinline__ float key_to_float(unsigned k) {
    return (k & 0x80000000u) ? __uint_as_float(k ^ 0x80000000u)
                             : __uint_as_float(~k);
}

__global__ __launch_bounds__(256) void mine_kernel(
    const float* __restrict__ negbuf,
    const float* __restrict__ regbuf,
    const int*   __restrict__ matched,
    const float* __restrict__ meta,
    float* __restrict__ pos, float* __restrict__ negs, float* __restrict__ regs)
{
    __shared__ unsigned s_key[DD];
    __shared__ float    s_red[256];
    __shared__ unsigned s_cnt;

    const int b = blockIdx.x, t = threadIdx.x;
    const float* __restrict__ ce = negbuf + (size_t)b * DD;
    const float* __restrict__ rg = regbuf + (size_t)b * DD;

    float psum = 0.0f, rsum = 0.0f;
    for (int d = t; d < DD; d += 256) {
        float v = ce[d];
        rsum += rg[d];
        unsigned k;
        if (matched[d]) { psum += v; k = 0u; }   // excluded from mining (ce>=0 => key>=0x80000000)
        else {
            unsigned u = __float_as_uint(v);
            k = (u & 0x80000000u) ? ~u : (u | 0x80000000u);
        }
        s_key[d] = k;
    }
    s_red[t] = psum; __syncthreads();
    for (int s = 128; s > 0; s >>= 1) { if (t < s) s_red[t] += s_red[t + s]; __syncthreads(); }
    if (t == 0) pos[b] = s_red[0];
    __syncthreads();
    s_red[t] = rsum; __syncthreads();
    for (int s = 128; s > 0; s >>= 1) { if (t < s) s_red[t] += s_red[t + s]; __syncthreads(); }
    if (t == 0) regs[b] = s_red[0];
    __syncthreads();

    const int N = ((const int*)meta)[1];
    long long K = (long long)NEG_RATIO * N;
    long long fin = (long long)DD - N;
    if (K > fin) K = fin;

    if (K > 0) {
        unsigned prefix = 0;
        for (int bit = 31; bit >= 0; --bit) {
            unsigned cand = prefix | (1u << bit);
            if (t == 0) s_cnt = 0;
            __syncthreads();
            unsigned local = 0;
            for (int d = t; d < DD; d += 256) local += (s_key[d] >= cand) ? 1u : 0u;
            atomicAdd(&s_cnt, local);
            __syncthreads();
            if ((long long)s_cnt >= K) prefix = cand;
            __syncthreads();
        }
        // sum strictly-greater values, fill ties with the kth value
        if (t == 0) s_cnt = 0;
        __syncthreads();
        float lsum = 0.0f; unsigned lgt = 0;
        for (int d = t; d < DD; d += 256) {
            unsigned k = s_key[d];
            if (k > prefix) { ++lgt; lsum += key_to_float(k); }
        }
        atomicAdd(&s_cnt, lgt);
        s_red[t] = lsum; __syncthreads();
        for (int s = 128; s > 0; s >>= 1) { if (t < s) s_red[t] += s_red[t + s]; __syncthreads(); }
        if (t == 0) {
            float kth = key_to_float(prefix);
            negs[b] = s_red[0] + (float)(K - (long long)s_cnt) * kth;
        }
    } else if (t == 0) {
        negs[b] = 0.0f;
    }
}

// ---------------------------------------------------------------------------
// Kernel 4: finalize the three scalars.
// ---------------------------------------------------------------------------
__global__ void final_kernel(const float* __restrict__ meta,
                             const float* __restrict__ pos,
                             const float* __restrict__ negs,
                             const float* __restrict__ regs,
                             float* __restrict__ out)
{
    float Nf = meta[0];
    float cls = 0.0f, loc = 0.0f;
    for (int b = 0; b < BB; ++b) {
        cls += pos[b] + negs[b];
        loc += regs[b];
    }
    cls = cls / Nf / (float)BB;
    loc = loc / Nf / (float)BB;
    out[0] = cls + loc;
    out[1] = loc;
    out[2] = cls;
}

// ---------------------------------------------------------------------------
extern "C" void kernel_launch(void* const* d_in, const int* in_sizes, int n_in,
                              void* d_out, int out_size, void* d_ws, size_t ws_size,
                              hipStream_t stream) {
    (void)in_sizes; (void)n_in; (void)out_size; (void)ws_size;
    const float* po = (const float*)d_in[0];   // predicted_offsets [B][D][4]
    const float* pc = (const float*)d_in[1];   // predicted_classes [B][D][C]
    const float* db = (const float*)d_in[2];   // default_boxes [D][4]
    const float* tg = (const float*)d_in[3];   // targets [B*A][6]
    float* out = (float*)d_out;

    auto align_up = [](size_t v) { return (v + 255) & ~(size_t)255; };
    char* w = (char*)d_ws;
    float* meta    = (float*)w;                 w += align_up(16);
    int*   labels  = (int*)w;                   w += align_up(sizeof(int) * DD);
    int*   matched = (int*)w;                   w += align_up(sizeof(int) * DD);
    int*   bwa     = (int*)w;                   w += align_up(sizeof(int) * DD);
    float* trueoff = (float*)w;                 w += align_up(sizeof(float) * DD * 4);
    float* negbuf  = (float*)w;                 w += align_up(sizeof(float) * (size_t)BB * DD);
    float* regbuf  = (float*)w;                 w += align_up(sizeof(float) * (size_t)BB * DD);
    float* pos     = (float*)w;                 w += align_up(sizeof(float) * BB);
    float* negs    = (float*)w;                 w += align_up(sizeof(float) * BB);
    float* regs    = (float*)w;                 w += align_up(sizeof(float) * BB);

    match_kernel<<<1, 256, 0, stream>>>(db, tg, labels, matched, bwa, trueoff, meta);

    const int nrows = BB * DD;                       // 558,848 (multiple of 128)
    ce_kernel<<<nrows / ROWS_PER_BLOCK, ROWS_PER_BLOCK, 0, stream>>>(
        pc, po, labels, matched, trueoff, negbuf, regbuf);

    mine_kernel<<<BB, 256, 0, stream>>>(negbuf, regbuf, matched, meta, pos, negs, regs);

    final_kernel<<<1, 1, 0, stream>>>(meta, pos, negs, regs, out);
}